// ModelNew_69758858821769
// MI455X (gfx1250) — compile-verified
//
#include <hip/hip_runtime.h>
#include <math.h>

typedef __attribute__((ext_vector_type(2))) float v2f;
typedef __attribute__((ext_vector_type(8))) float v8f;
typedef __attribute__((ext_vector_type(4))) unsigned int v4u;
typedef __attribute__((ext_vector_type(4))) int v4i;
typedef __attribute__((ext_vector_type(8))) int v8i;

#define IN_F            512
#define OUT_F           32
#define WAVES           8
#define BLOCK           (WAVES * 32)
#define ROWS_PER_WAVE   16
#define ROWS_PER_BLOCK  (WAVES * ROWS_PER_WAVE)   // 128
#define CK              64                        // K columns staged per chunk
#define XSTRIDE         (CK + 4)                  // TDM pad: 4 dw every 64 dw -> stride%64==4
#define WSTRIDE         (IN_F + 4)                // TDM pad: 2 dw every 256 dw -> stride%64==4
#define NCHUNK          (IN_F / CK)               // 8

// --- Tensor Data Mover: 2D tile (global -> LDS) with LDS padding -------------
// D# packing per CDNA5 ISA:
//   group0: [0]=count=1, [1]=lds byte addr, [2]=global_addr[31:0],
//           [3]=global_addr[56:32] | type(=2)<<30
//   group1: [0]=data_size(4B)=2<<16 | pad_enable<<20 | pad_interval<<22 | pad_amount<<25
//           bits[79:48]=tensor_dim0, [111:80]=tensor_dim1, [127:112]=tile_dim0,
//           [143:128]=tile_dim1, [159:144]=tile_dim2(0), [207:160]=tensor_dim0_stride
//   groups2/3: zeros (2D tensor, iterate/gather disabled)
__device__ __forceinline__ void tdm_load_2d_to_lds(
    unsigned lds_byte_off, const void* gsrc,
    unsigned tile_cols, unsigned tile_rows, unsigned tensor_dim1,
    unsigned row_stride_elems, unsigned pad_interval_code, unsigned pad_amount_code)
{
  unsigned long long ga = (unsigned long long)(uintptr_t)gsrc;
  v4u g0;
  g0.x = 1u;                                                     // count=1, user mode
  g0.y = lds_byte_off;                                           // LDS byte address
  g0.z = (unsigned)ga;                                           // global_addr[31:0]
  g0.w = (unsigned)((ga >> 32) & 0x01FFFFFFull) | (2u << 30);    // addr[56:32] | type=2
  v8i g1;
  g1[0] = (int)((2u << 16)                       // data_size = 4 bytes
              | (1u << 20)                       // pad_enable
              | (pad_interval_code << 22)        // pad every 2^(code+1) dwords
              | (pad_amount_code << 25));        // pad (code+1) dwords
  g1[1] = (int)(((unsigned)IN_F & 0xFFFFu) << 16);                       // tensor_dim0 lo
  g1[2] = (int)((((unsigned)IN_F >> 16) & 0xFFFFu) | ((tensor_dim1 & 0xFFFFu) << 16));
  g1[3] = (int)(((tensor_dim1 >> 16) & 0xFFFFu) | ((tile_cols & 0xFFFFu) << 16)); // tile_dim0
  g1[4] = (int)(tile_rows & 0xFFFFu);            // tile_dim1 (tile_dim2 = 0)
  g1[5] = (int)row_stride_elems;                 // tensor_dim0_stride[31:0]
  g1[6] = 0;                                     // stride0[47:32] | stride1[15:0]
  g1[7] = 0;                                     // stride1[47:16]
  v4i gz = {0, 0, 0, 0};
  asm volatile("" ::: "memory");
#if __clang_major__ >= 23
  v8i gz8 = {0, 0, 0, 0, 0, 0, 0, 0};
  __builtin_amdgcn_tensor_load_to_lds(g0, g1, gz, gz, gz8, 0);   // clang-23 6-arg form
#else
  __builtin_amdgcn_tensor_load_to_lds(g0, g1, gz, gz, 0);        // ROCm 7.2 5-arg form
#endif
  asm volatile("" ::: "memory");
}

// Online logsumexp merge of (max, sumexp) pairs packed into one 64-bit word.
__device__ __forceinline__ void merge_pair(unsigned long long* p, float mx, float sm) {
  unsigned long long assumed, old = *p;
  do {
    assumed = old;
    float omx = __uint_as_float((unsigned int)(assumed & 0xffffffffull));
    float osm = __uint_as_float((unsigned int)(assumed >> 32));
    float nm  = fmaxf(omx, mx);
    float ns  = osm * expf(omx - nm) + sm * expf(mx - nm);
    unsigned long long nv = (unsigned long long)__float_as_uint(nm)
                          | ((unsigned long long)__float_as_uint(ns) << 32);
    old = atomicCAS(p, assumed, nv);
  } while (old != assumed);
}

__global__ __launch_bounds__(BLOCK) void gemm_max_lse_kernel(
    const float* __restrict__ x, const float* __restrict__ W,
    unsigned long long* __restrict__ pair)
{
  __shared__ float wlds[OUT_F * WSTRIDE];                   // 32*516 f = 66.0 KB
  __shared__ float xbuf[WAVES][2][ROWS_PER_WAVE * XSTRIDE]; // 8*2*1088 f = 69.6 KB
  __shared__ float mrow[ROWS_PER_BLOCK];
  __shared__ float red[ROWS_PER_BLOCK];

  const int tid  = threadIdx.x;
  const int lane = tid & 31;
  const int wid  = tid >> 5;
  const int r    = lane & 15;   // A: M (batch row in tile); B: N (output in tile)
  const int h    = lane >> 4;   // lane-half selects K pair {0,1} vs {2,3}

  // Stage all of W (32x512) into padded LDS with ONE TDM op per block.
  // pad 2 dwords every 256 -> physical row stride 516 dwords (bank-conflict-free).
  if (wid == 0) {
    tdm_load_2d_to_lds((unsigned)(uintptr_t)&wlds[0], W,
                       IN_F, OUT_F, OUT_F, IN_F,
                       /*pad_interval: 256 dw*/ 7, /*pad_amount: 2 dw*/ 1);
    __builtin_amdgcn_s_wait_tensorcnt((short)0);
  }
  __syncthreads();

  const size_t rowbase = (size_t)(blockIdx.x * WAVES + wid) * ROWS_PER_WAVE;
  const float* xrow = x + rowbase * IN_F;

  // Per-wave double-buffered x staging: one TDM per 16x64 tile.
  // pad 4 dwords every 64 -> physical row stride 68 dwords (bank-conflict-free).
  auto tdm_load_x = [&](int ch) {
    tdm_load_2d_to_lds((unsigned)(uintptr_t)&xbuf[wid][ch & 1][0],
                       xrow + (size_t)ch * CK,
                       CK, ROWS_PER_WAVE, ROWS_PER_WAVE, IN_F,
                       /*pad_interval: 64 dw*/ 5, /*pad_amount: 4 dw*/ 3);
  };

  v8f c0 = {0.f, 0.f, 0.f, 0.f, 0.f, 0.f, 0.f, 0.f};
  v8f c1 = c0;

  auto compute_chunk = [&](int ch) {
    // W column c = ch*CK + kk + 2h; +2 dwords once past the 256-dword pad point.
    const float* xb = &xbuf[wid][ch & 1][0] + r * XSTRIDE + 2 * h;
    const float* w0 = &wlds[r * WSTRIDE] + ch * CK + 2 * h + (ch >= 4 ? 2 : 0);
    const float* w1 = w0 + 16 * WSTRIDE;
    #pragma unroll
    for (int kk = 0; kk < CK; kk += 4) {
      v2f a  = *(const v2f*)(xb + kk);   // A frag (16x4 f32), conflict-free ds_load_b64
      v2f b0 = *(const v2f*)(w0 + kk);   // B frag tile0 (outputs 0..15)
      v2f b1 = *(const v2f*)(w1 + kk);   // B frag tile1 (outputs 16..31)
      c0 = __builtin_amdgcn_wmma_f32_16x16x4_f32(false, a, false, b0, (short)0, c0, false, false);
      c1 = __builtin_amdgcn_wmma_f32_16x16x4_f32(false, a, false, b1, (short)0, c1, false, false);
    }
  };

  // TDM pipeline: issue ch+1, wait for <=1 outstanding (ch done, in-order), compute ch.
  tdm_load_x(0);
  for (int ch = 0; ch < NCHUNK - 1; ++ch) {
    tdm_load_x(ch + 1);
    __builtin_amdgcn_s_wait_tensorcnt((short)1);
    asm volatile("" ::: "memory");
    compute_chunk(ch);
  }
  __builtin_amdgcn_s_wait_tensorcnt((short)0);
  asm volatile("" ::: "memory");
  compute_chunk(NCHUNK - 1);

  // Max over N=0..31 per row. c0/c1 lane layout: VGPR i holds M=i (lanes 0-15)
  // and M=8+i (lanes 16-31), N = lane&15 (tile0) / 16+(lane&15) (tile1).
  float t[8];
  #pragma unroll
  for (int i = 0; i < 8; ++i) t[i] = fmaxf(c0[i], c1[i]);
  #pragma unroll
  for (int off = 8; off >= 1; off >>= 1) {
    #pragma unroll
    for (int i = 0; i < 8; ++i)
      t[i] = fmaxf(t[i], __shfl_xor(t[i], off, 32));  // stays within 16-lane halves
  }
  if ((lane & 15) == 0) {               // lane 0: rows 0..7, lane 16: rows 8..15
    int base = wid * 16 + h * 8;
    #pragma unroll
    for (int i = 0; i < 8; ++i) mrow[base + i] = t[i];
  }
  __syncthreads();

  // Block-level logsumexp partial over the 128 row-maxes.
  float m = (tid < ROWS_PER_BLOCK) ? mrow[tid] : -INFINITY;
  if (tid < ROWS_PER_BLOCK) red[tid] = m;
  __syncthreads();
  for (int s = ROWS_PER_BLOCK / 2; s > 0; s >>= 1) {
    if (tid < s) red[tid] = fmaxf(red[tid], red[tid + s]);
    __syncthreads();
  }
  float bmax = red[0];
  __syncthreads();
  float e = (tid < ROWS_PER_BLOCK) ? expf(m - bmax) : 0.0f;
  if (tid < ROWS_PER_BLOCK) red[tid] = e;
  __syncthreads();
  for (int s = ROWS_PER_BLOCK / 2; s > 0; s >>= 1) {
    if (tid < s) red[tid] += red[tid + s];
    __syncthreads();
  }
  if (tid == 0) merge_pair(pair, bmax, red[0]);
}

__global__ void init_pair_kernel(unsigned long long* p) {
  // max = -inf (low 32 bits), sumexp = 0.0f (high 32 bits)
  *p = (unsigned long long)__float_as_uint(-INFINITY);
}

__global__ void broadcast_kernel(const unsigned long long* __restrict__ p,
                                 float* __restrict__ out, int n) {
  unsigned long long v = *p;
  float mx = __uint_as_float((unsigned int)(v & 0xffffffffull));
  float sm = __uint_as_float((unsigned int)(v >> 32));
  float l2 = mx + logf(sm) + logf((float)n);   // lse(m) + log(B)
  int i = blockIdx.x * blockDim.x + threadIdx.x;
  if (i < n) out[i] = l2;
}

extern "C" void kernel_launch(void* const* d_in, const int* in_sizes, int n_in,
                              void* d_out, int out_size, void* d_ws, size_t ws_size,
                              hipStream_t stream) {
  const float* x = (const float*)d_in[0];
  const float* W = (const float*)d_in[1];
  float* out = (float*)d_out;
  unsigned long long* pair = (unsigned long long*)d_ws;

  const int batch = in_sizes[0] / IN_F;          // 524288
  init_pair_kernel<<<1, 1, 0, stream>>>(pair);   // must re-init every call (graph replay)
  const int nblocks = batch / ROWS_PER_BLOCK;    // 4096
  gemm_max_lse_kernel<<<nblocks, BLOCK, 0, stream>>>(x, W, pair);
  broadcast_kernel<<<(out_size + 255) / 256, 256, 0, stream>>>(pair, out, batch);
}